// IRMatchingModule_21912923144748
// MI455X (gfx1250) — compile-verified
//
#include <hip/hip_runtime.h>
#include <hip/hip_bf16.h>
#include <math.h>

// Problem constants (from reference setup_inputs):
// B=8, S=512, H=768, Q=64, D=256, C=2
#define BB 8
#define SS 512
#define HH 768
#define QQ 64
#define DD 256
#define H2 1536   // 2*H
#define MPAD 16   // M padded to 16 for WMMA

typedef __attribute__((ext_vector_type(2))) float v2f;
typedef __attribute__((ext_vector_type(8))) float v8f;

__device__ __forceinline__ float wave_reduce_add(float v) {
    v += __shfl_xor(v, 16, 32);
    v += __shfl_xor(v, 8, 32);
    v += __shfl_xor(v, 4, 32);
    v += __shfl_xor(v, 2, 32);
    v += __shfl_xor(v, 1, 32);
    return v;
}

// ---------------------------------------------------------------------------
// Kernel 1: per-batch fused prep. Produces feat[16 x 1536]:
//   feat[b][0:768]      = query_avg
//   feat[b][768:1536]   = query_based_doc  (via exact algebraic collapse)
//   feat[8+b][:]        = 0 (padding rows for the M=16 WMMA tile)
// ---------------------------------------------------------------------------
__global__ __launch_bounds__(256) void prep_kernel(
    const float* __restrict__ features,   // (B,S,H)
    const int*   __restrict__ idxA,       // (B,Q)
    const int*   __restrict__ idxB,       // (B,D)
    float*       __restrict__ feat)       // (16,1536)
{
    __shared__ int   s_aidx[QQ];
    __shared__ float s_amask[QQ];
    __shared__ float s_qinv[QQ];
    __shared__ int   s_bidx[DD];
    __shared__ float s_bmask[DD];
    __shared__ float s_t[DD];
    __shared__ float s_u[HH];
    __shared__ float s_len[2];

    const int b    = blockIdx.x;
    const int tid  = threadIdx.x;
    const int lane = tid & 31;
    const int wave = tid >> 5;
    const float* fb = features + (size_t)b * SS * HH;

    // ---- load indices + masks --------------------------------------------
    if (tid < QQ) {
        int ix = idxA[b * QQ + tid];
        s_aidx[tid]  = ix;
        s_amask[tid] = (ix > 0) ? 1.0f : 0.0f;
    }
    {
        int ix = idxB[b * DD + tid];
        s_bidx[tid]  = ix;
        s_bmask[tid] = (ix > 0) ? 1.0f : 0.0f;
    }
    __syncthreads();

    // ---- lengths (tiny serial reductions on two lanes) -------------------
    if (tid == 0)  { float s = 0.f; for (int q = 0; q < QQ; ++q) s += s_amask[q]; s_len[0] = s; }
    if (tid == 32) { float s = 0.f; for (int d = 0; d < DD; ++d) s += s_bmask[d]; s_len[1] = s; }

    // ---- per-q inverse norms: 8 rows per wave ----------------------------
    for (int i = 0; i < QQ / 8; ++i) {
        int q = wave * (QQ / 8) + i;
        const float* row = fb + (size_t)s_aidx[q] * HH;
        float s2 = 0.f;
        for (int h = lane; h < HH; h += 32) { float v = row[h]; s2 += v * v; }
        s2 = wave_reduce_add(s2);
        if (lane == 0) s_qinv[q] = 1.0f / fmaxf(sqrtf(s2), 1e-8f);
    }
    __syncthreads();

    const float inv_alen = 1.0f / s_len[0];
    const float inv_blen = 1.0f / s_len[1];

    // ---- u[h] = sum_q (amask/a_len)*qn[q,h];  query_avg ------------------
    for (int h = tid; h < HH; h += 256) {
        float sa = 0.f, su = 0.f;
        for (int q = 0; q < QQ; ++q) {
            float m = s_amask[q];            // uniform across threads
            if (m != 0.f) {
                float v = fb[(size_t)s_aidx[q] * HH + h];
                sa += v;
                su += v * s_qinv[q];
            }
        }
        s_u[h] = su * inv_alen;
        feat[(size_t)b * H2 + h] = sa * inv_alen;   // query_avg
    }
    // zero WMMA padding row (8+b)
    for (int h = tid; h < H2; h += 256) feat[(size_t)(BB + b) * H2 + h] = 0.f;
    __syncthreads();

    // ---- t[d] = bmask[d] * (u . dn[d]) / b_len : 32 rows per wave --------
    for (int i = 0; i < DD / 8; ++i) {
        int d = wave * (DD / 8) + i;
        float m = s_bmask[d];                // uniform across the wave
        if (m == 0.f) { if (lane == 0) s_t[d] = 0.f; continue; }
        const float* row = fb + (size_t)s_bidx[d] * HH;
        float n2 = 0.f, dot = 0.f;
        for (int h = lane; h < HH; h += 32) {
            float v = row[h];
            n2  += v * v;
            dot += v * s_u[h];
        }
        n2  = wave_reduce_add(n2);
        dot = wave_reduce_add(dot);
        if (lane == 0)
            s_t[d] = (dot / fmaxf(sqrtf(n2), 1e-8f)) * inv_blen;
    }
    __syncthreads();

    // ---- query_based_doc[h] = sum_d t[d] * d_vec[d,h] --------------------
    for (int h = tid; h < HH; h += 256) {
        float acc = 0.f;
        for (int d = 0; d < DD; ++d) {
            float td = s_t[d];
            if (td != 0.f)
                acc += td * fb[(size_t)s_bidx[d] * HH + h];
        }
        feat[(size_t)b * H2 + HH + h] = acc;
    }
}

// ---------------------------------------------------------------------------
// Kernel 2: hidden(16x768) = relu(feat(16x1536) @ W1(1536x768) + b1)
// One wave per 16x16 output tile, V_WMMA_F32_16X16X4_F32, K swept in steps
// of 4 (384 WMMAs/tile). Full f32 precision.
//
// VGPR layouts (ISA 7.12.2, wave32):
//   A 16x4 : lanes 0-15 -> M=lane, K={k0,k0+1}; lanes 16-31 -> K={k0+2,k0+3}
//   B 4x16 : lanes 0-15 -> N=lane, K rows {k0,k0+1}; lanes 16-31 -> {k0+2,k0+3}
//   C/D    : VGPR v -> row v (lanes 0-15) / row v+8 (lanes 16-31), col = lane&15
// ---------------------------------------------------------------------------
__global__ __launch_bounds__(256) void mlp1_wmma_kernel(
    const float* __restrict__ feat,    // (16,1536)
    const float* __restrict__ W1,      // (1536,768) row-major
    const float* __restrict__ b1,      // (768)
    float*       __restrict__ hidden)  // (16,768)
{
    const int lane  = threadIdx.x & 31;
    const int wave  = threadIdx.x >> 5;
    const int tile  = blockIdx.x * 8 + wave;          // 0..47 N-tiles
    const int nsub  = lane & 15;
    const int ncol  = tile * 16 + nsub;
    const int koff  = (lane >> 4) << 1;               // 0 or 2

    const float* ap = feat + (size_t)nsub * H2 + koff;
    const float* bp = W1 + (size_t)koff * HH + ncol;

    v8f c = {};
    for (int k0 = 0; k0 < H2; k0 += 4) {
        float2 av = *(const float2*)(ap + k0);        // even offset: 8B aligned
        v2f a; a.x = av.x; a.y = av.y;
        v2f bv;
        bv.x = bp[(size_t)k0 * HH];
        bv.y = bp[(size_t)(k0 + 1) * HH];
        c = __builtin_amdgcn_wmma_f32_16x16x4_f32(
                /*neg_a=*/false, a, /*neg_b=*/false, bv,
                /*c_mod=*/(short)0, c, /*reuse_a=*/false, /*reuse_b=*/false);
    }

    const int   mbase = (lane >> 4) << 3;             // 0 or 8
    const float bias  = b1[ncol];
    #pragma unroll
    for (int v = 0; v < 8; ++v) {
        int m = mbase + v;
        hidden[(size_t)m * HH + ncol] = fmaxf(c[v] + bias, 0.0f);
    }
}

// ---------------------------------------------------------------------------
// Kernel 3: logits(8x2) = hidden(8x768) @ W2(768x2) + b2
// One wave per (batch, class) output.
// ---------------------------------------------------------------------------
__global__ __launch_bounds__(32) void logits_kernel(
    const float* __restrict__ hidden,  // (16,768), rows 0..7 valid
    const float* __restrict__ W2,      // (768,2)
    const float* __restrict__ b2,      // (2)
    float*       __restrict__ out)     // (8,2)
{
    const int o    = blockIdx.x;       // 0..15
    const int b    = o >> 1;
    const int cls  = o & 1;
    const int lane = threadIdx.x;
    const float* hp = hidden + (size_t)b * HH;
    float s = 0.f;
    for (int i = lane; i < HH; i += 32) s += hp[i] * W2[i * 2 + cls];
    s = wave_reduce_add(s);
    if (lane == 0) out[b * 2 + cls] = s + b2[cls];
}

// ---------------------------------------------------------------------------
extern "C" void kernel_launch(void* const* d_in, const int* in_sizes, int n_in,
                              void* d_out, int out_size, void* d_ws, size_t ws_size,
                              hipStream_t stream) {
    const float* features = (const float*)d_in[0];
    const int*   idxA     = (const int*)d_in[1];
    const int*   idxB     = (const int*)d_in[2];
    const float* W1       = (const float*)d_in[3];
    const float* b1       = (const float*)d_in[4];
    const float* W2       = (const float*)d_in[5];
    const float* b2       = (const float*)d_in[6];
    float*       out      = (float*)d_out;

    float* ws     = (float*)d_ws;
    float* feat   = ws;                        // 16*1536 floats
    float* hidden = ws + (size_t)MPAD * H2;    // 16*768 floats

    prep_kernel<<<dim3(BB), dim3(256), 0, stream>>>(features, idxA, idxB, feat);
    mlp1_wmma_kernel<<<dim3(6), dim3(256), 0, stream>>>(feat, W1, b1, hidden);
    logits_kernel<<<dim3(16), dim3(32), 0, stream>>>(hidden, W2, b2, out);
}